// _MambaMixer_33449205301613
// MI455X (gfx1250) — compile-verified
//
#include <hip/hip_runtime.h>
#include <hip/hip_bf16.h>
#include <cstddef>

// ---------------- problem constants ----------------
constexpr int D_MODEL  = 1024;
constexpr int D_STATE  = 16;
constexpr int D_CONV   = 4;
constexpr int D_INNER  = 2048;      // EXPAND * D_MODEL
constexpr int DT_RANK  = 64;
constexpr int BATCH    = 2;
constexpr int SEQLEN   = 2048;
constexpr int BL       = BATCH * SEQLEN;      // 4096 rows for all GEMMs
constexpr int PROJ_W   = DT_RANK + 2 * D_STATE; // 96
constexpr int NCHUNK   = 16;
constexpr int CLEN     = SEQLEN / NCHUNK;     // 128

typedef __attribute__((ext_vector_type(2))) float v2f;
typedef __attribute__((ext_vector_type(4))) float v4f;
typedef __attribute__((ext_vector_type(8))) float v8f;

static __device__ __forceinline__ v2f lo2(v4f v) { return (v2f){v[0], v[1]}; }
static __device__ __forceinline__ v2f hi2(v4f v) { return (v2f){v[2], v[3]}; }

// =====================================================================
// Generic fp32 WMMA GEMM:   C[M,N] = A[M,K] @ W[N,K]^T   (+ epilogue)
// One wave computes a 16 x (16*NTILES) tile using V_WMMA_F32_16X16X4_F32.
//
// Fragment layout (ISA 7.12.2): lanes 0-15 supply logical k={0,1},
// lanes 16-31 supply logical k={2,3}.  K-accumulation is order-free, so
// we process 8 columns per step with ONE b128 load per matrix:
//   lane<16  loads cols [kb .. kb+3], lane>=16 loads cols [kb+4 .. kb+7]
//   WMMA #1 (elements 0,1) covers global cols {kb,kb+1,kb+4,kb+5}
//   WMMA #2 (elements 2,3) covers global cols {kb+2,kb+3,kb+6,kb+7}
// -> 1+NTILES b128 loads feed 2*NTILES WMMAs per step.
// The kb-loop is software-pipelined one step deep.
// EPI: 0 = plain store, 1 = softplus(x + bias[n])
// =====================================================================
template <int NTILES, int EPI>
__global__ void wmma_gemm(const float* __restrict__ A,
                          const float* __restrict__ W,
                          float* __restrict__ C,
                          const float* __restrict__ bias,
                          int M, int N, int K,
                          int lda, int ldb, int ldc)
{
    const int wave = (blockIdx.x * blockDim.x + threadIdx.x) >> 5;
    const int lane = threadIdx.x & 31;
    const int tiles_n = N / (16 * NTILES);
    const int mt = wave / tiles_n;
    const int nt = wave % tiles_n;

    const int m0 = mt * 16;
    const int n0 = nt * 16 * NTILES;

    const int half = lane >> 4;          // 0 or 1
    const int l15  = lane & 15;
    const int koff = half * 4;           // lane<16 -> cols kb.., lane>=16 -> kb+4..

    const float* ap = A + (size_t)(m0 + l15) * lda + koff;

    const float* wp[NTILES];
#pragma unroll
    for (int j = 0; j < NTILES; ++j)
        wp[j] = W + (size_t)(n0 + j * 16 + l15) * ldb + koff;

    v8f acc[NTILES];
#pragma unroll
    for (int j = 0; j < NTILES; ++j) acc[j] = (v8f){0,0,0,0,0,0,0,0};

    // ---- software pipeline: prefetch fragments for kb=0 ----
    v4f a_cur = *(const v4f*)(ap);
    v4f b_cur[NTILES];
#pragma unroll
    for (int j = 0; j < NTILES; ++j) b_cur[j] = *(const v4f*)(wp[j]);

#pragma unroll 2
    for (int kb = 8; kb < K; kb += 8) {
        // issue next-step loads before consuming current fragments
        v4f a_nxt = *(const v4f*)(ap + kb);
        v4f b_nxt[NTILES];
#pragma unroll
        for (int j = 0; j < NTILES; ++j) b_nxt[j] = *(const v4f*)(wp[j] + kb);

#pragma unroll
        for (int j = 0; j < NTILES; ++j)
            acc[j] = __builtin_amdgcn_wmma_f32_16x16x4_f32(
                false, lo2(a_cur), false, lo2(b_cur[j]), (short)0, acc[j], false, false);
#pragma unroll
        for (int j = 0; j < NTILES; ++j)
            acc[j] = __builtin_amdgcn_wmma_f32_16x16x4_f32(
                false, hi2(a_cur), false, hi2(b_cur[j]), (short)0, acc[j], false, false);

        a_cur = a_nxt;
#pragma unroll
        for (int j = 0; j < NTILES; ++j) b_cur[j] = b_nxt[j];
    }
    // ---- drain last step ----
#pragma unroll
    for (int j = 0; j < NTILES; ++j)
        acc[j] = __builtin_amdgcn_wmma_f32_16x16x4_f32(
            false, lo2(a_cur), false, lo2(b_cur[j]), (short)0, acc[j], false, false);
#pragma unroll
    for (int j = 0; j < NTILES; ++j)
        acc[j] = __builtin_amdgcn_wmma_f32_16x16x4_f32(
            false, hi2(a_cur), false, hi2(b_cur[j]), (short)0, acc[j], false, false);

    const int crow0 = m0 + half * 8;
#pragma unroll
    for (int j = 0; j < NTILES; ++j) {
        const int cn = n0 + j * 16 + l15;
        float bv = 0.f;
        if constexpr (EPI == 1) bv = bias[cn];
#pragma unroll
        for (int g = 0; g < 8; ++g) {
            float v = acc[j][g];
            if constexpr (EPI == 1) {
                float x = v + bv;                      // softplus(x + dt_proj_b)
                v = (x > 20.f) ? x : log1pf(__expf(x));
            }
            C[(size_t)(crow0 + g) * ldc + cn] = v;
        }
    }
}

// =====================================================================
// Depthwise causal conv (k=4) + bias + SiLU.  x lives in first half of xz.
// =====================================================================
__global__ void conv_silu(const float* __restrict__ xz,
                          const float* __restrict__ cw,
                          const float* __restrict__ cb,
                          float* __restrict__ xfeat)
{
    const int i = blockIdx.x * blockDim.x + threadIdx.x;   // over BL*D_INNER
    const int d  = i % D_INNER;
    const int bl = i / D_INNER;
    const int l  = bl % SEQLEN;

    float acc = cb[d];
#pragma unroll
    for (int j = 0; j < D_CONV; ++j) {
        const int ll = l - (D_CONV - 1) + j;
        if (ll >= 0)
            acc += xz[(size_t)(bl - l + ll) * (2 * D_INNER) + d] * cw[d * D_CONV + j];
    }
    xfeat[(size_t)i] = acc / (1.f + __expf(-acc));          // silu
}

// =====================================================================
// Chunked selective scan.
// Phase 1: per (b,d,chunk) lane: local scan with zero init ->
//          chunk dA-product P[16] and local tail state t[16].
// =====================================================================
__global__ void scan_p1(const float* __restrict__ dtp,
                        const float* __restrict__ xf,
                        const float* __restrict__ proj,
                        const float* __restrict__ A_log,
                        float* __restrict__ Pw,
                        float* __restrict__ Tw)
{
    const int gid = blockIdx.x * blockDim.x + threadIdx.x; // B*NCHUNK*D_INNER
    const int d = gid % D_INNER;
    const int c = (gid / D_INNER) % NCHUNK;
    const int b = gid / (D_INNER * NCHUNK);

    float Av[D_STATE];
#pragma unroll
    for (int q = 0; q < 4; ++q) {
        float4 v = ((const float4*)(A_log + d * D_STATE))[q];
        Av[4*q+0] = -__expf(v.x); Av[4*q+1] = -__expf(v.y);
        Av[4*q+2] = -__expf(v.z); Av[4*q+3] = -__expf(v.w);
    }

    float t[D_STATE], P[D_STATE];
#pragma unroll
    for (int n = 0; n < D_STATE; ++n) { t[n] = 0.f; P[n] = 1.f; }

    const int bl0 = b * SEQLEN + c * CLEN;
    for (int s = 0; s < CLEN; ++s) {
        const int bl = bl0 + s;
        const size_t off = (size_t)bl * D_INNER + d;
        const float dtv = dtp[off];
        const float du  = dtv * xf[off];

        float Bv[D_STATE];
        const float* bp = proj + (size_t)bl * PROJ_W + DT_RANK;
#pragma unroll
        for (int q = 0; q < 4; ++q) {
            float4 v = ((const float4*)bp)[q];
            Bv[4*q+0]=v.x; Bv[4*q+1]=v.y; Bv[4*q+2]=v.z; Bv[4*q+3]=v.w;
        }
#pragma unroll
        for (int n = 0; n < D_STATE; ++n) {
            float a = __expf(dtv * Av[n]);
            P[n] *= a;
            t[n] = fmaf(a, t[n], du * Bv[n]);
        }
    }

    const size_t o = (((size_t)b * D_INNER + d) * NCHUNK + c) * D_STATE;
#pragma unroll
    for (int q = 0; q < 4; ++q) {
        ((float4*)(Pw + o))[q] = make_float4(P[4*q], P[4*q+1], P[4*q+2], P[4*q+3]);
        ((float4*)(Tw + o))[q] = make_float4(t[4*q], t[4*q+1], t[4*q+2], t[4*q+3]);
    }
}

// Phase 2: tiny sequential scan over chunks per (b,d,n) -> initial states S0.
__global__ void scan_p2(const float* __restrict__ Pw,
                        const float* __restrict__ Tw,
                        float* __restrict__ S0)
{
    const int gid = blockIdx.x * blockDim.x + threadIdx.x;  // B*D_INNER*16
    const size_t base = (size_t)(gid / D_STATE) * (NCHUNK * D_STATE) + (gid % D_STATE);
    float s = 0.f;
    for (int c = 0; c < NCHUNK; ++c) {
        const size_t o = base + (size_t)c * D_STATE;
        S0[o] = s;
        s = Pw[o] * s + Tw[o];
    }
}

// Phase 3: re-run each chunk with correct initial state; fuse y = s·C + u*D,
// then y *= silu(z).
__global__ void scan_p3(const float* __restrict__ dtp,
                        const float* __restrict__ xf,
                        const float* __restrict__ proj,
                        const float* __restrict__ A_log,
                        const float* __restrict__ S0,
                        const float* __restrict__ xz,
                        const float* __restrict__ Dp,
                        float* __restrict__ y)
{
    const int gid = blockIdx.x * blockDim.x + threadIdx.x;
    const int d = gid % D_INNER;
    const int c = (gid / D_INNER) % NCHUNK;
    const int b = gid / (D_INNER * NCHUNK);

    float Av[D_STATE];
#pragma unroll
    for (int q = 0; q < 4; ++q) {
        float4 v = ((const float4*)(A_log + d * D_STATE))[q];
        Av[4*q+0] = -__expf(v.x); Av[4*q+1] = -__expf(v.y);
        Av[4*q+2] = -__expf(v.z); Av[4*q+3] = -__expf(v.w);
    }

    float st[D_STATE];
    {
        const size_t o = (((size_t)b * D_INNER + d) * NCHUNK + c) * D_STATE;
#pragma unroll
        for (int q = 0; q < 4; ++q) {
            float4 v = ((const float4*)(S0 + o))[q];
            st[4*q+0]=v.x; st[4*q+1]=v.y; st[4*q+2]=v.z; st[4*q+3]=v.w;
        }
    }
    const float Dv = Dp[d];

    const int bl0 = b * SEQLEN + c * CLEN;
    for (int s = 0; s < CLEN; ++s) {
        const int bl = bl0 + s;
        const size_t off = (size_t)bl * D_INNER + d;
        const float dtv = dtp[off];
        const float uv  = xf[off];
        const float du  = dtv * uv;

        float Bv[D_STATE], Cv[D_STATE];
        const float* bp = proj + (size_t)bl * PROJ_W + DT_RANK;
        const float* cp = bp + D_STATE;
#pragma unroll
        for (int q = 0; q < 4; ++q) {
            float4 v = ((const float4*)bp)[q];
            Bv[4*q+0]=v.x; Bv[4*q+1]=v.y; Bv[4*q+2]=v.z; Bv[4*q+3]=v.w;
            float4 w = ((const float4*)cp)[q];
            Cv[4*q+0]=w.x; Cv[4*q+1]=w.y; Cv[4*q+2]=w.z; Cv[4*q+3]=w.w;
        }

        float acc = 0.f;
#pragma unroll
        for (int n = 0; n < D_STATE; ++n) {
            float a = __expf(dtv * Av[n]);
            st[n] = fmaf(a, st[n], du * Bv[n]);
            acc = fmaf(st[n], Cv[n], acc);
        }
        acc = fmaf(uv, Dv, acc);
        const float zv = xz[(size_t)bl * (2 * D_INNER) + D_INNER + d];
        acc *= zv / (1.f + __expf(-zv));     // * silu(z)
        y[off] = acc;
    }
}

// =====================================================================
// Host-side pipeline
// =====================================================================
extern "C" void kernel_launch(void* const* d_in, const int* in_sizes, int n_in,
                              void* d_out, int out_size, void* d_ws, size_t ws_size,
                              hipStream_t stream)
{
    const float* inputs    = (const float*)d_in[0];  // [B,L,D_MODEL]
    const float* in_proj_w = (const float*)d_in[1];  // [2*D_INNER, D_MODEL]
    const float* conv_w    = (const float*)d_in[2];  // [D_INNER,1,4]
    const float* conv_b    = (const float*)d_in[3];  // [D_INNER]
    const float* x_proj_w  = (const float*)d_in[4];  // [96, D_INNER]
    const float* dt_proj_w = (const float*)d_in[5];  // [D_INNER, 64]
    const float* dt_proj_b = (const float*)d_in[6];  // [D_INNER]
    const float* A_log     = (const float*)d_in[7];  // [D_INNER, 16]
    const float* Dparam    = (const float*)d_in[8];  // [D_INNER]
    const float* out_proj_w= (const float*)d_in[9];  // [D_MODEL, D_INNER]
    float* out = (float*)d_out;

    float* ws    = (float*)d_ws;
    float* xz    = ws;                                  // BL * 4096
    float* xfeat = xz    + (size_t)BL * 2 * D_INNER;    // BL * 2048
    float* proj  = xfeat + (size_t)BL * D_INNER;        // BL * 96
    float* dtp   = proj  + (size_t)BL * PROJ_W;         // BL * 2048
    float* yb    = dtp   + (size_t)BL * D_INNER;        // BL * 2048
    float* Pw    = yb    + (size_t)BL * D_INNER;        // B*D_INNER*NCHUNK*16
    float* Tw    = Pw    + (size_t)BATCH * D_INNER * NCHUNK * D_STATE;
    float* S0    = Tw    + (size_t)BATCH * D_INNER * NCHUNK * D_STATE;

    // 1) xz = inputs @ in_proj_w.T      M=4096 N=4096 K=1024
    {
        const int waves = (BL / 16) * (2 * D_INNER / 64);       // 16384
        wmma_gemm<4, 0><<<waves / 4, 128, 0, stream>>>(
            inputs, in_proj_w, xz, nullptr,
            BL, 2 * D_INNER, D_MODEL, D_MODEL, D_MODEL, 2 * D_INNER);
    }
    // 2) depthwise causal conv + bias + silu -> x_feat
    {
        const int total = BL * D_INNER;                          // 8.39M
        conv_silu<<<total / 256, 256, 0, stream>>>(xz, conv_w, conv_b, xfeat);
    }
    // 3) proj = x_feat @ x_proj_w.T     M=4096 N=96 K=2048 (one wave = all 6 N-tiles)
    {
        const int waves = BL / 16;                               // 256
        wmma_gemm<6, 0><<<waves / 4, 128, 0, stream>>>(
            xfeat, x_proj_w, proj, nullptr,
            BL, PROJ_W, D_INNER, D_INNER, D_INNER, PROJ_W);
    }
    // 4) dtp = softplus(proj[:, :64] @ dt_proj_w.T + dt_proj_b)   M=4096 N=2048 K=64
    {
        const int waves = (BL / 16) * (D_INNER / 64);            // 8192
        wmma_gemm<4, 1><<<waves / 4, 128, 0, stream>>>(
            proj, dt_proj_w, dtp, dt_proj_b,
            BL, D_INNER, DT_RANK, PROJ_W, DT_RANK, D_INNER);
    }
    // 5-7) chunked selective scan
    {
        const int lanes = BATCH * NCHUNK * D_INNER;              // 65536
        scan_p1<<<lanes / 256, 256, 0, stream>>>(dtp, xfeat, proj, A_log, Pw, Tw);
        const int lanes2 = BATCH * D_INNER * D_STATE;            // 65536
        scan_p2<<<lanes2 / 256, 256, 0, stream>>>(Pw, Tw, S0);
        scan_p3<<<lanes / 256, 256, 0, stream>>>(dtp, xfeat, proj, A_log, S0,
                                                 xz, Dparam, yb);
    }
    // 8) out = y @ out_proj_w.T         M=4096 N=1024 K=2048
    {
        const int waves = (BL / 16) * (D_MODEL / 64);            // 4096
        wmma_gemm<4, 0><<<waves / 4, 128, 0, stream>>>(
            yb, out_proj_w, out, nullptr,
            BL, D_MODEL, D_INNER, D_INNER, D_INNER, D_MODEL);
    }
}